// WeightedClassificationLoss_36232344109307
// MI455X (gfx1250) — compile-verified
//
#include <hip/hip_runtime.h>

typedef float v2f __attribute__((ext_vector_type(2)));
typedef float v8f __attribute__((ext_vector_type(8)));

#define BLOCK 256
#define GRID  2048
#define WAVES_PER_BLOCK (BLOCK / 32)

// ---------------------------------------------------------------------------
// Pass 1: grid-stride streaming pass. Fully branchless per-row math so the
// body is a straight v_cndmask/v_max/v_exp chain; wave32 reduction via
// V_WMMA_F32_16X16X4_F32; per-block partial written to d_ws.
// ---------------------------------------------------------------------------
__global__ void wcl_main(const float4* __restrict__ logits,
                         const unsigned int* __restrict__ targets_lo, // int64 low words, stride 2
                         const float* __restrict__ weights,
                         float* __restrict__ partial,
                         int B) {
    __shared__ float sW[16];
    __shared__ float sWave[WAVES_PER_BLOCK];

    if (threadIdx.x < 16) sW[threadIdx.x] = weights[threadIdx.x];
    __syncthreads();

    const int stride = GRID * BLOCK;
    float acc = 0.0f;

    for (int i = blockIdx.x * BLOCK + threadIdx.x; i < B; i += stride) {
        float4 x = logits[i];                         // global_load_b128
        unsigned int t = targets_lo[2 * i];           // global_load_b32 (low word of int64)

        // ---- branchless argmax (first-max semantics via strict >) ----
        float m01 = fmaxf(x.x, x.y);
        float m23 = fmaxf(x.z, x.w);
        int   p01 = (x.y > x.x) ? 1 : 0;              // v_cmp + v_cndmask
        int   p23 = (x.w > x.z) ? 3 : 2;
        float best = fmaxf(m01, m23);
        int   pred = (m23 > m01) ? p23 : p01;

        // ---- log-softmax CE at target ----
        float e0 = __expf(x.x - best);
        float e1 = __expf(x.y - best);
        float e2 = __expf(x.z - best);
        float e3 = __expf(x.w - best);
        float sum = (e0 + e1) + (e2 + e3);

        // branchless select of x[t] via bit tests -> cndmask tree
        float lo = (t & 1u) ? x.y : x.x;
        float hi = (t & 1u) ? x.w : x.z;
        float xt = (t & 2u) ? hi : lo;

        float ce  = __logf(sum) - xt + best;
        float sev = sW[(t << 2) | (unsigned)pred];    // ds_load_b32
        acc += sev + ce;
    }

    // ---- wave32 reduction with one WMMA ----
    // A (16x4 f32, 2 VGPRs): VGPR0 = acc, VGPR1 = 0
    //   -> A[l%16][l<16?0:2] = acc_l, K=1,3 columns zero
    // B (4x16 f32, 2 VGPRs): all ones
    // D[m][n] = acc_m + acc_{m+16}; lanes 0-15 hold rows 0-7 (8 VGPRs),
    // lanes 16-31 hold rows 8-15. Sum own 8 values, combine lane 0 + lane 16.
    v2f a; a[0] = acc;  a[1] = 0.0f;
    v2f b; b[0] = 1.0f; b[1] = 1.0f;
    v8f c = {0.0f, 0.0f, 0.0f, 0.0f, 0.0f, 0.0f, 0.0f, 0.0f};
    c = __builtin_amdgcn_wmma_f32_16x16x4_f32(false, a, false, b,
                                              (short)0, c, false, false);
    float s = ((c[0] + c[1]) + (c[2] + c[3])) + ((c[4] + c[5]) + (c[6] + c[7]));
    float waveSum = __shfl(s, 0, 32) + __shfl(s, 16, 32);

    const int lane = threadIdx.x & 31;
    const int wid  = threadIdx.x >> 5;
    if (lane == 0) sWave[wid] = waveSum;
    __syncthreads();

    if (threadIdx.x == 0) {
        float bs = 0.0f;
        #pragma unroll
        for (int j = 0; j < WAVES_PER_BLOCK; ++j) bs += sWave[j];
        partial[blockIdx.x] = bs;
    }
}

// ---------------------------------------------------------------------------
// Pass 2: deterministic final reduction of GRID partials, scale by 1/B.
// ---------------------------------------------------------------------------
__global__ void wcl_final(const float* __restrict__ partial,
                          float* __restrict__ out,
                          int n, float invB) {
    __shared__ float sWave[WAVES_PER_BLOCK];
    float v = 0.0f;
    for (int i = threadIdx.x; i < n; i += BLOCK) v += partial[i];

    #pragma unroll
    for (int off = 16; off > 0; off >>= 1)
        v += __shfl_xor(v, off, 32);

    const int lane = threadIdx.x & 31;
    const int wid  = threadIdx.x >> 5;
    if (lane == 0) sWave[wid] = v;
    __syncthreads();

    if (threadIdx.x == 0) {
        float total = 0.0f;
        #pragma unroll
        for (int j = 0; j < WAVES_PER_BLOCK; ++j) total += sWave[j];
        out[0] = total * invB;
    }
}

extern "C" void kernel_launch(void* const* d_in, const int* in_sizes, int n_in,
                              void* d_out, int out_size, void* d_ws, size_t ws_size,
                              hipStream_t stream) {
    const float* outputs            = (const float*)d_in[0];
    const unsigned int* targets_lo  = (const unsigned int*)d_in[1]; // int64 viewed as u32 pairs
    const float* weights            = (const float*)d_in[2];
    float* out     = (float*)d_out;
    float* partial = (float*)d_ws;   // GRID floats = 8 KB of scratch

    const int B = in_sizes[1];       // number of rows (targets element count)

    wcl_main<<<GRID, BLOCK, 0, stream>>>((const float4*)outputs, targets_lo,
                                         weights, partial, B);
    wcl_final<<<1, BLOCK, 0, stream>>>(partial, out, GRID, 1.0f / (float)B);
}